// SelfAttentiveLBLBiLM_86878598464080
// MI455X (gfx1250) — compile-verified
//
#include <hip/hip_runtime.h>

// Problem constants (match reference)
#define B_ 8
#define S_ 512
#define D_ 512
#define H_ 64
#define DK_ 8
#define WIDTH_ 16
#define N_ (S_ + 2 * WIDTH_)   // 544
#define NL_ 2
#define WIN_ (WIDTH_ + 2)      // 18 keys max per query
#define BN_ (B_ * N_)          // 4352 rows (padded)
#define BS_ (B_ * S_)          // 4096 rows (sliced)
static const size_t BND_ = (size_t)BN_ * D_;  // 2,228,224
static const size_t BSD_ = (size_t)BS_ * D_;  // 2,097,152

typedef __attribute__((ext_vector_type(2))) float v2f;
typedef __attribute__((ext_vector_type(8))) float v8f;

// ---------------------------------------------------------------------------
// Build padded input: xp[b, p, d] = left_pad | inputs | right_pad
// ---------------------------------------------------------------------------
__global__ __launch_bounds__(256)
void pad_kernel(const float* __restrict__ x, const float* __restrict__ lp,
                const float* __restrict__ rp, float* __restrict__ xp) {
  size_t idx = (size_t)blockIdx.x * blockDim.x + threadIdx.x;
  if (idx >= BND_) return;
  int dd = (int)(idx % D_);
  size_t bn = idx / D_;
  int p = (int)(bn % N_);
  int b = (int)(bn / N_);
  float v;
  if (p < WIDTH_)            v = lp[p * D_ + dd];
  else if (p < WIDTH_ + S_)  v = x[((size_t)b * S_ + (p - WIDTH_)) * D_ + dd];
  else                       v = rp[(p - WIDTH_ - S_) * D_ + dd];
  xp[idx] = v;
}

// ---------------------------------------------------------------------------
// C[M,Nout] = A[M,512] @ W[Nout,512]^T + bias   (torch Linear semantics)
// Block = 8 waves; all waves share one 16-row A tile, staged in LDS once
// (coalesced b128 global loads, padded stride to avoid bank conflicts).
// Wave computes a 16x64 tile: one A-frag (from LDS), four B-frags (global),
// 4 WMMAs per K-step, with explicit register double-buffering so the next
// step's loads are in flight underneath the current step's WMMAs.
// f32 WMMA 16x16x4 lane layout (ISA 7.12.2):
//   A: lane holds {M = lane&15, K = kbase, kbase+1}, kbase = 2*(lane>>4)
//   B: lane holds {N = lane&15, K = kbase, kbase+1}  (B[k][n] = W[n][k])
//   C: VGPR r = row (r + 8*(lane>>4)), col = lane&15
// ---------------------------------------------------------------------------
__global__ __launch_bounds__(256)
void gemm_wmma_f32(const float* __restrict__ A, int lda,
                   const float* __restrict__ W, int ldw,
                   const float* __restrict__ bias,
                   float* __restrict__ C, int ldc,
                   int M, int Nout) {
  constexpr int KK = 512;
  constexpr int LDA_S = KK + 8;           // padded LDS row stride (floats)
  __shared__ float As[16 * LDA_S];        // ~33 KB

  const int lane = threadIdx.x;           // 0..31
  const int wave = threadIdx.y;           // 0..7
  const int tid  = wave * 32 + lane;      // 0..255
  const int row0 = blockIdx.y * 16;
  const int col0 = (blockIdx.x * 8 + wave) * 64;

  // --- Stage the block's 16 x 512 A tile into LDS (coalesced) ---
  {
    constexpr int NVEC = 16 * (KK / 4);   // 2048 float4
#pragma unroll
    for (int v = 0; v < NVEC / 256; ++v) {
      int e   = tid + v * 256;
      int row = e >> 7;                   // 128 float4 per row
      int cv  = e & 127;
      int rg  = row0 + row; if (rg > M - 1) rg = M - 1;
      float4 val = *(const float4*)(A + (size_t)rg * lda + cv * 4);
      *(float4*)&As[row * LDA_S + cv * 4] = val;
    }
  }
  __syncthreads();

  if (row0 >= M || col0 >= Nout) return;  // wave-uniform (never taken here)

  const int ml = lane & 15;               // M index (A) / N index (B)
  const int kb = (lane >> 4) << 1;        // K sub-offset: 0 or 2

  const float* w0 = W + (size_t)(col0 + ml) * ldw + kb;
  const float* w1 = w0 + (size_t)16 * ldw;
  const float* w2 = w0 + (size_t)32 * ldw;
  const float* w3 = w0 + (size_t)48 * ldw;
  const float* arow = &As[ml * LDA_S + kb];

  v8f acc0 = {}, acc1 = {}, acc2 = {}, acc3 = {};

  // Prologue: fragments for k = 0
  v2f a  = *(const v2f*)(arow);
  v2f b0 = *(const v2f*)(w0);
  v2f b1 = *(const v2f*)(w1);
  v2f b2 = *(const v2f*)(w2);
  v2f b3 = *(const v2f*)(w3);

#pragma unroll 4
  for (int k = 4; k < KK; k += 4) {
    // Issue next step's loads first so they overlap the WMMAs below.
    v2f an  = *(const v2f*)(arow + k);
    v2f b0n = *(const v2f*)(w0 + k);
    v2f b1n = *(const v2f*)(w1 + k);
    v2f b2n = *(const v2f*)(w2 + k);
    v2f b3n = *(const v2f*)(w3 + k);
    acc0 = __builtin_amdgcn_wmma_f32_16x16x4_f32(false, a, false, b0, (short)0, acc0, false, false);
    acc1 = __builtin_amdgcn_wmma_f32_16x16x4_f32(false, a, false, b1, (short)0, acc1, false, false);
    acc2 = __builtin_amdgcn_wmma_f32_16x16x4_f32(false, a, false, b2, (short)0, acc2, false, false);
    acc3 = __builtin_amdgcn_wmma_f32_16x16x4_f32(false, a, false, b3, (short)0, acc3, false, false);
    a = an; b0 = b0n; b1 = b1n; b2 = b2n; b3 = b3n;
  }
  // Epilogue: last K-step
  acc0 = __builtin_amdgcn_wmma_f32_16x16x4_f32(false, a, false, b0, (short)0, acc0, false, false);
  acc1 = __builtin_amdgcn_wmma_f32_16x16x4_f32(false, a, false, b1, (short)0, acc1, false, false);
  acc2 = __builtin_amdgcn_wmma_f32_16x16x4_f32(false, a, false, b2, (short)0, acc2, false, false);
  acc3 = __builtin_amdgcn_wmma_f32_16x16x4_f32(false, a, false, b3, (short)0, acc3, false, false);

  const int mbase = row0 + ((lane >> 4) << 3);
  float bv0 = bias[col0 +  0 + ml];
  float bv1 = bias[col0 + 16 + ml];
  float bv2 = bias[col0 + 32 + ml];
  float bv3 = bias[col0 + 48 + ml];
#pragma unroll
  for (int r = 0; r < 8; ++r) {
    float* crow = C + (size_t)(mbase + r) * ldc + col0 + ml;
    crow[ 0] = acc0[r] + bv0;
    crow[16] = acc1[r] + bv1;
    crow[32] = acc2[r] + bv2;
    crow[48] = acc3[r] + bv3;
  }
}

// ---------------------------------------------------------------------------
// Banded-window attention, head dim 8, window <= 18 keys. One thread/(b,i,h),
// with h fastest-varying so 64 consecutive threads cover one contiguous
// 2KB qkv row (fully coalesced 32B/thread loads).
// dir==0: keys j in [i-17, i]; dir==1: keys j in [i, i+17].
// ---------------------------------------------------------------------------
__global__ __launch_bounds__(256)
void attn_kernel(const float* __restrict__ qkv, float* __restrict__ out, int dir) {
  int idx = blockIdx.x * blockDim.x + threadIdx.x;
  if (idx >= B_ * H_ * N_) return;
  int h    = idx % H_;
  int rest = idx / H_;
  int i    = rest % N_;
  int b    = rest / N_;

  const size_t base = (size_t)(b * N_ + i) * D_ + h * DK_;
  const float* qp = qkv + base;
  const float* kbase = qkv + BND_ + (size_t)(b * N_) * D_ + h * DK_;
  const float* vbase = qkv + 2 * BND_ + (size_t)(b * N_) * D_ + h * DK_;

  float q[DK_];
  *(float4*)&q[0] = *(const float4*)(qp);
  *(float4*)&q[4] = *(const float4*)(qp + 4);

  const int j0 = (dir == 0) ? ((i - WIDTH_ - 1 > 0) ? i - WIDTH_ - 1 : 0) : i;
  const int j1 = (dir == 0) ? i : ((i + WIDTH_ + 1 < N_ - 1) ? i + WIDTH_ + 1 : N_ - 1);
  const float scale = 0.35355339059327373f;  // 1/sqrt(8)

  float sv[WIN_];
  float mx = -3.0e38f;
#pragma unroll
  for (int t = 0; t < WIN_; ++t) {
    int j = j0 + t;
    float sc = -3.0e38f;
    if (j <= j1) {
      const float* kp = kbase + (size_t)j * D_;
      float kr[DK_];
      *(float4*)&kr[0] = *(const float4*)(kp);
      *(float4*)&kr[4] = *(const float4*)(kp + 4);
      float dot = 0.f;
#pragma unroll
      for (int d = 0; d < DK_; ++d) dot += q[d] * kr[d];
      sc = dot * scale;
    }
    sv[t] = sc;
    mx = fmaxf(mx, sc);
  }
  float sum = 0.f;
#pragma unroll
  for (int t = 0; t < WIN_; ++t) {
    float p = __expf(sv[t] - mx);  // padded entries -> exp(-huge) == 0
    sv[t] = p;
    sum += p;
  }
  const float inv = 1.f / sum;
  float o[DK_] = {};
#pragma unroll
  for (int t = 0; t < WIN_; ++t) {
    int j = j0 + t;
    if (j <= j1) {
      const float* vp = vbase + (size_t)j * D_;
      float vr[DK_];
      *(float4*)&vr[0] = *(const float4*)(vp);
      *(float4*)&vr[4] = *(const float4*)(vp + 4);
      float w = sv[t] * inv;
#pragma unroll
      for (int d = 0; d < DK_; ++d) o[d] += w * vr[d];
    }
  }
  float* op = out + base;
  float4 o0; o0.x = o[0]; o0.y = o[1]; o0.z = o[2]; o0.w = o[3];
  float4 o1; o1.x = o[4]; o1.y = o[5]; o1.z = o[6]; o1.w = o[7];
  *(float4*)(op)     = o0;
  *(float4*)(op + 4) = o1;
}

// ---------------------------------------------------------------------------
// Extract center slice [WIDTH : WIDTH+S] into contiguous [B*S, D]
// ---------------------------------------------------------------------------
__global__ __launch_bounds__(256)
void slice_kernel(const float* __restrict__ proj, float* __restrict__ hx) {
  size_t idx = (size_t)blockIdx.x * blockDim.x + threadIdx.x;
  if (idx >= BSD_) return;
  int dd = (int)(idx % D_);
  size_t ms = idx / D_;
  int s = (int)(ms % S_);
  int b = (int)(ms / S_);
  hx[idx] = proj[((size_t)(b * N_ + WIDTH_ + s)) * D_ + dd];
}

// ---------------------------------------------------------------------------
// Highway gate: hx = sig(proj[:,D:]) * hx + (1-sig) * relu(proj[:,:D])
// ---------------------------------------------------------------------------
__global__ __launch_bounds__(256)
void gate_kernel(const float* __restrict__ hp, float* __restrict__ hx) {
  size_t idx = (size_t)blockIdx.x * blockDim.x + threadIdx.x;
  if (idx >= BSD_) return;
  int dd = (int)(idx % D_);
  size_t m = idx / D_;
  float lin = hp[m * (2 * D_) + dd];
  float gin = hp[m * (2 * D_) + D_ + dd];
  float g  = 1.f / (1.f + __expf(-gin));
  float nl = fmaxf(lin, 0.f);
  hx[idx] = g * hx[idx] + (1.f - g) * nl;
}

// ---------------------------------------------------------------------------
// Write direction's columns into d_out, duplicated (all_layers, last)
// ---------------------------------------------------------------------------
__global__ __launch_bounds__(256)
void outw_kernel(const float* __restrict__ hx, float* __restrict__ out, int dir) {
  size_t idx = (size_t)blockIdx.x * blockDim.x + threadIdx.x;
  if (idx >= BSD_) return;
  int dd = (int)(idx % D_);
  size_t m = idx / D_;
  size_t o = m * (size_t)(2 * D_) + (size_t)dir * D_ + dd;
  float v = hx[idx];
  out[o] = v;
  out[o + (size_t)BS_ * 2 * D_] = v;
}

// ---------------------------------------------------------------------------
extern "C" void kernel_launch(void* const* d_in, const int* in_sizes, int n_in,
                              void* d_out, int out_size, void* d_ws, size_t ws_size,
                              hipStream_t stream) {
  const float* inputs    = (const float*)d_in[0];
  const float* left_pad  = (const float*)d_in[1];
  const float* right_pad = (const float*)d_in[2];
  const float* Wqkv[2] = {(const float*)d_in[3],  (const float*)d_in[7]};
  const float* bqkv[2] = {(const float*)d_in[4],  (const float*)d_in[8]};
  const float* Wo[2]   = {(const float*)d_in[5],  (const float*)d_in[9]};
  const float* bo[2]   = {(const float*)d_in[6],  (const float*)d_in[10]};
  const float* hwW[2]  = {(const float*)d_in[11], (const float*)d_in[13]};
  const float* hwb[2]  = {(const float*)d_in[12], (const float*)d_in[14]};
  float* out = (float*)d_out;

  // Workspace layout (~53 MB; proj/hproj alias the dead qkv region)
  float* ws    = (float*)d_ws;
  float* xpad  = ws;                 // BND_
  float* qkv   = xpad + BND_;        // 3*BND_
  float* attn  = qkv + 3 * BND_;     // BND_
  float* hx    = attn + BND_;        // BSD_
  float* proj  = qkv;                // alias: qkv dead after attention
  float* hproj = qkv + BND_;         // alias: BS_*1024 fits in 2*BND_

  dim3 blk(32, 8);  // 8 wave32 per block

  { int n = (int)BND_; pad_kernel<<<(n + 255) / 256, 256, 0, stream>>>(inputs, left_pad, right_pad, xpad); }

  for (int dir = 0; dir < 2; ++dir) {
    // QKV projections: qkv[t] = xpad @ Wqkv[t]^T + bqkv[t]
    for (int t = 0; t < 3; ++t) {
      dim3 grd(1, BN_ / 16);
      gemm_wmma_f32<<<grd, blk, 0, stream>>>(
          xpad, D_, Wqkv[dir] + (size_t)t * D_ * D_, D_, bqkv[dir] + t * D_,
          qkv + (size_t)t * BND_, D_, BN_, D_);
    }
    // Windowed attention
    { int n = B_ * H_ * N_;
      attn_kernel<<<(n + 255) / 256, 256, 0, stream>>>(qkv, attn, dir); }
    // Output projection
    { dim3 grd(1, BN_ / 16);
      gemm_wmma_f32<<<grd, blk, 0, stream>>>(attn, D_, Wo[dir], D_, bo[dir],
                                             proj, D_, BN_, D_); }
    // Slice center [WIDTH : WIDTH+S]
    { int n = (int)BSD_; slice_kernel<<<(n + 255) / 256, 256, 0, stream>>>(proj, hx); }
    // Highway layers
    for (int l = 0; l < NL_; ++l) {
      dim3 grd(2, BS_ / 16);
      gemm_wmma_f32<<<grd, blk, 0, stream>>>(
          hx, D_, hwW[dir] + (size_t)l * 2 * D_ * D_, D_, hwb[dir] + l * 2 * D_,
          hproj, 2 * D_, BS_, 2 * D_);
      int n = (int)BSD_;
      gate_kernel<<<(n + 255) / 256, 256, 0, stream>>>(hproj, hx);
    }
    // Emit this direction's half of the concatenated output (twice)
    { int n = (int)BSD_; outw_kernel<<<(n + 255) / 256, 256, 0, stream>>>(hx, out, dir); }
  }
}